// DynConv2d_57458072486036
// MI455X (gfx1250) — compile-verified
//
#include <hip/hip_runtime.h>
#include <hip/hip_bf16.h>

typedef float v2f __attribute__((ext_vector_type(2)));
typedef float v4f __attribute__((ext_vector_type(4)));
typedef float v8f __attribute__((ext_vector_type(8)));

#define NB 2
#define NC 64
#define NP 8192
#define NK 16
#define NPTS (NB * NP)           // 16384
#define NWAVES_CONV 2048
#define NELEM_BN (NB * NP * NK)  // 262144

static __device__ __forceinline__ v8f wmma_f32(v2f a, v2f b, v8f c) {
    // D(16x16) = A(16x4) * B(4x16) + C, full fp32
    return __builtin_amdgcn_wmma_f32_16x16x4_f32(
        /*neg_a=*/false, a, /*neg_b=*/false, b,
        /*c_mod=*/(short)0, c, /*reuse_a=*/false, /*reuse_b=*/false);
}

// ---------------------------------------------------------------------------
// Kernel 1: transpose x (B,C,N) -> ft (B*N, C) contiguous, and squared norms.
// ---------------------------------------------------------------------------
__global__ __launch_bounds__(256) void prep_kernel(
    const float* __restrict__ x, float* __restrict__ ft, float* __restrict__ sq) {
    int p = blockIdx.x * 256 + threadIdx.x;   // 0..16383
    int b = p >> 13, n = p & (NP - 1);
    const float* xb = x + (size_t)b * NC * NP;
    float s = 0.f;
    float* dst = ft + (size_t)p * NC;
#pragma unroll
    for (int c = 0; c < NC; ++c) {
        float v = xb[(size_t)c * NP + n];
        dst[c] = v;
        s += v * v;
    }
    sq[p] = s;
}

// ---------------------------------------------------------------------------
// Kernel 2: fused distance-GEMM + top-16. One wave handles 16 queries.
// A = candidate tile (M), B = query tile (N): lane holds one query column.
// Double-buffered A staging; sq in LDS (DScnt-decoupled from global loads).
// ---------------------------------------------------------------------------
__global__ __launch_bounds__(128) void knn_kernel(
    const float* __restrict__ ft, const float* __restrict__ sq, int* __restrict__ nn) {
    __shared__ __align__(16) float lsq[NP];  // 32 KB, one batch per block

    const int lane = threadIdx.x & 31;
    const int lo = lane & 15, hi = lane >> 4;
    const int wave = blockIdx.x * 4 + (threadIdx.x >> 5);  // 0..1023
    const int b = wave >> 9;                               // uniform per block
    const int qbase = (wave & 511) << 4;
    const float* ftb = ft + (size_t)b * NP * NC;
    const float* sqb = sq + (size_t)b * NP;

    for (int i = threadIdx.x * 4; i < NP; i += 128 * 4)
        *(v4f*)&lsq[i] = *(const v4f*)&sqb[i];
    __syncthreads();

    // preload B operand (query features), channels 4cc+2hi..+1 of query qbase+lo
    v2f bq[16];
#pragma unroll
    for (int cc = 0; cc < 16; ++cc)
        bq[cc] = *(const v2f*)(ftb + (size_t)(qbase + lo) * NC + 4 * cc + 2 * hi);

    float bd[16];
    int bi[16];
#pragma unroll
    for (int j = 0; j < 16; ++j) { bd[j] = 3.0e38f; bi[j] = 0; }
    float wval = 3.0e38f;
    int wslot = 0;

    auto insert = [&](float d, int idx) {
        if (d < wval) {
#pragma unroll
            for (int j = 0; j < 16; ++j)
                if (j == wslot) { bd[j] = d; bi[j] = idx; }
            wval = bd[0]; wslot = 0;
#pragma unroll
            for (int j = 1; j < 16; ++j)
                if (bd[j] > wval) { wval = bd[j]; wslot = j; }
        }
    };

    auto loadA = [&](int mb, v2f* a) {
        const float* src = ftb + (size_t)((mb & (NP - 1)) + lo) * NC + 2 * hi;
#pragma unroll
        for (int cc = 0; cc < 16; ++cc)
            a[cc] = *(const v2f*)(src + 4 * cc);
    };

    auto tile = [&](int mb, const v2f* a) {
        v8f acc = {0.f, 0.f, 0.f, 0.f, 0.f, 0.f, 0.f, 0.f};
#pragma unroll
        for (int cc = 0; cc < 16; ++cc)
            acc = wmma_f32(a[cc], bq[cc], acc);
        // lane owns candidates mb+8*hi .. +7 for query qbase+lo.
        // rank-invariant distance: |c|^2 - 2<c,q>  (query norm dropped)
        int cb = mb + 8 * hi;
        v4f s0 = *(const v4f*)&lsq[cb];
        v4f s1 = *(const v4f*)&lsq[cb + 4];
        float d[8];
#pragma unroll
        for (int r = 0; r < 8; ++r) {
            float sc = (r < 4) ? s0[r & 3] : s1[r & 3];
            d[r] = fmaf(-2.0f, acc[r], sc);
        }
        float tmin = fminf(fminf(fminf(d[0], d[1]), fminf(d[2], d[3])),
                           fminf(fminf(d[4], d[5]), fminf(d[6], d[7])));
        if (tmin < wval) {
#pragma unroll
            for (int r = 0; r < 8; ++r) insert(d[r], cb + r);
        }
    };

    v2f aA[16], aB[16];
    loadA(0, aA);
    for (int mb = 0; mb < NP; mb += 32) {
        loadA(mb + 16, aB);   // in flight while aA is consumed
        tile(mb, aA);
        loadA(mb + 32, aA);   // in flight while aB is consumed
        tile(mb + 16, aB);
    }

    // merge high-half lane's 16 into low-half lane's 16
#pragma unroll
    for (int j = 0; j < 16; ++j) {
        float dj = __shfl_down(bd[j], 16, 32);
        int ij = __shfl_down(bi[j], 16, 32);
        insert(dj, ij);
    }
    if (lane < 16) {
        int q = qbase + lane;
        int* dst = nn + ((size_t)b * NP + q) * NK;
#pragma unroll
        for (int j = 0; j < 16; ++j) dst[j] = bi[j];
    }
}

// ---------------------------------------------------------------------------
// Kernel 3: edge-conv GEMM, accumulate per-wave partial sum / sumsq of y.
// ---------------------------------------------------------------------------
__global__ __launch_bounds__(256) void stats_kernel(
    const float* __restrict__ ft, const int* __restrict__ nn,
    const float* __restrict__ W, float* __restrict__ part) {
    __shared__ __align__(16) float lW[64 * 128];
    for (int i = threadIdx.x * 4; i < 64 * 128; i += 256 * 4)
        *(v4f*)&lW[i] = *(const v4f*)&W[i];
    __syncthreads();

    const int lane = threadIdx.x & 31;
    const int lo = lane & 15, hi = lane >> 4;
    const int wid = blockIdx.x * 8 + (threadIdx.x >> 5);  // 0..2047

    float sacc[32], qacc[32];
#pragma unroll
    for (int i = 0; i < 32; ++i) { sacc[i] = 0.f; qacc[i] = 0.f; }

    for (int p = wid; p < NPTS; p += NWAVES_CONV) {
        int b = p >> 13, n = p & (NP - 1);
        const float* ftb = ft + (size_t)b * NP * NC;
        int j = nn[(size_t)p * NK + lo];
        v2f fx[16], fd[16];
#pragma unroll
        for (int cc = 0; cc < 16; ++cc) {
            v2f xi = *(const v2f*)(ftb + (size_t)n * NC + 4 * cc + 2 * hi);
            v2f xj = *(const v2f*)(ftb + (size_t)j * NC + 4 * cc + 2 * hi);
            fx[cc] = xi;
            fd[cc] = xj - xi;
        }
#pragma unroll
        for (int ot = 0; ot < 4; ++ot) {
            const float* wrow = &lW[(16 * ot + lo) * 128 + 2 * hi];
            v8f y = {0.f, 0.f, 0.f, 0.f, 0.f, 0.f, 0.f, 0.f};
#pragma unroll
            for (int h = 0; h < 2; ++h) {
                v2f aW[16];
#pragma unroll
                for (int cc = 0; cc < 16; ++cc)
                    aW[cc] = *(const v2f*)(wrow + 4 * (16 * h + cc));
#pragma unroll
                for (int cc = 0; cc < 16; ++cc)
                    y = wmma_f32(aW[cc], (h == 0) ? fx[cc] : fd[cc], y);
            }
#pragma unroll
            for (int r = 0; r < 8; ++r) {
                float v = y[r];
                sacc[ot * 8 + r] += v;
                qacc[ot * 8 + r] += v * v;
            }
        }
    }
    // reduce over the 16 neighbor-lanes (lane bits 0..3)
#pragma unroll
    for (int i = 0; i < 32; ++i) {
#pragma unroll
        for (int m = 1; m < 16; m <<= 1) {
            sacc[i] += __shfl_xor(sacc[i], m, 32);
            qacc[i] += __shfl_xor(qacc[i], m, 32);
        }
    }
    if (lo == 0) {  // lanes 0 and 16
        float* pw = part + (size_t)wid * 128;
#pragma unroll
        for (int ot = 0; ot < 4; ++ot)
#pragma unroll
            for (int r = 0; r < 8; ++r) {
                int o = 16 * ot + 8 * hi + r;
                pw[o] = sacc[ot * 8 + r];
                pw[64 + o] = qacc[ot * 8 + r];
            }
    }
}

// ---------------------------------------------------------------------------
// Kernel 4: fold partials -> per-channel interleaved (scale, shift) pairs.
// ---------------------------------------------------------------------------
__global__ __launch_bounds__(64) void reduce_kernel(
    const float* __restrict__ part, const float* __restrict__ gamma,
    const float* __restrict__ beta, float* __restrict__ ss) {
    int o = threadIdx.x;
    double S = 0.0, Q = 0.0;
    for (int w = 0; w < NWAVES_CONV; ++w) {
        S += (double)part[(size_t)w * 128 + o];
        Q += (double)part[(size_t)w * 128 + 64 + o];
    }
    const double invM = 1.0 / (double)NELEM_BN;
    float mean = (float)(S * invM);
    float var = (float)(Q * invM) - mean * mean;
    float sc = gamma[o] * rsqrtf(var + 1e-5f);
    ss[2 * o] = sc;
    ss[2 * o + 1] = beta[o] - mean * sc;
}

// ---------------------------------------------------------------------------
// Kernel 5: recompute y, fused BN affine + LeakyReLU + max over K, write out.
// ---------------------------------------------------------------------------
__global__ __launch_bounds__(256) void final_kernel(
    const float* __restrict__ ft, const int* __restrict__ nn,
    const float* __restrict__ W, const float* __restrict__ ss,
    float* __restrict__ out) {
    __shared__ __align__(16) float lW[64 * 128];
    __shared__ __align__(16) float lss[128];
    for (int i = threadIdx.x * 4; i < 64 * 128; i += 256 * 4)
        *(v4f*)&lW[i] = *(const v4f*)&W[i];
    if (threadIdx.x < 128) lss[threadIdx.x] = ss[threadIdx.x];
    __syncthreads();

    const int lane = threadIdx.x & 31;
    const int lo = lane & 15, hi = lane >> 4;
    const int wid = blockIdx.x * 8 + (threadIdx.x >> 5);

    for (int p = wid; p < NPTS; p += NWAVES_CONV) {
        int b = p >> 13, n = p & (NP - 1);
        const float* ftb = ft + (size_t)b * NP * NC;
        int j = nn[(size_t)p * NK + lo];
        v2f fx[16], fd[16];
#pragma unroll
        for (int cc = 0; cc < 16; ++cc) {
            v2f xi = *(const v2f*)(ftb + (size_t)n * NC + 4 * cc + 2 * hi);
            v2f xj = *(const v2f*)(ftb + (size_t)j * NC + 4 * cc + 2 * hi);
            fx[cc] = xi;
            fd[cc] = xj - xi;
        }
        float mx[32];
#pragma unroll
        for (int ot = 0; ot < 4; ++ot) {
            const float* wrow = &lW[(16 * ot + lo) * 128 + 2 * hi];
            v8f y = {0.f, 0.f, 0.f, 0.f, 0.f, 0.f, 0.f, 0.f};
#pragma unroll
            for (int h = 0; h < 2; ++h) {
                v2f aW[16];
#pragma unroll
                for (int cc = 0; cc < 16; ++cc)
                    aW[cc] = *(const v2f*)(wrow + 4 * (16 * h + cc));
#pragma unroll
                for (int cc = 0; cc < 16; ++cc)
                    y = wmma_f32(aW[cc], (h == 0) ? fx[cc] : fd[cc], y);
            }
#pragma unroll
            for (int r = 0; r < 8; ++r) {
                int o = 16 * ot + 8 * hi + r;
                v2f st = *(const v2f*)&lss[2 * o];  // (scale, shift)
                float v = y[r] * st[0] + st[1];
                mx[ot * 8 + r] = (v >= 0.f) ? v : 0.2f * v;
            }
        }
        // max over the 16 neighbor lanes
#pragma unroll
        for (int i = 0; i < 32; ++i) {
#pragma unroll
            for (int m = 1; m < 16; m <<= 1) {
                float o2 = __shfl_xor(mx[i], m, 32);
                mx[i] = fmaxf(mx[i], o2);
            }
        }
        if (lo == 0) {
#pragma unroll
            for (int ot = 0; ot < 4; ++ot)
#pragma unroll
                for (int r = 0; r < 8; ++r) {
                    int o = 16 * ot + 8 * hi + r;
                    out[((size_t)b * 64 + o) * NP + n] = mx[ot * 8 + r];
                }
        }
    }
}

// ---------------------------------------------------------------------------
extern "C" void kernel_launch(void* const* d_in, const int* in_sizes, int n_in,
                              void* d_out, int out_size, void* d_ws, size_t ws_size,
                              hipStream_t stream) {
    const float* x = (const float*)d_in[0];      // (2,64,8192,1)
    const float* W = (const float*)d_in[1];      // (64,128)
    const float* gamma = (const float*)d_in[2];  // (64)
    const float* beta = (const float*)d_in[3];   // (64)
    float* out = (float*)d_out;                  // (2,64,8192)

    // workspace carve-up
    float* ft = (float*)d_ws;                         // 2*8192*64 floats (4 MB)
    float* sq = ft + (size_t)NPTS * NC;               // 16384 floats
    int* nn = (int*)(sq + NPTS);                      // 2*8192*16 ints (1 MB)
    float* part = (float*)(nn + (size_t)NPTS * NK);   // 2048*128 floats (1 MB)
    float* ss = part + (size_t)NWAVES_CONV * 128;     // 128 floats

    prep_kernel<<<NPTS / 256, 256, 0, stream>>>(x, ft, sq);
    knn_kernel<<<256, 128, 0, stream>>>(ft, sq, nn);
    stats_kernel<<<256, 256, 0, stream>>>(ft, nn, W, part);
    reduce_kernel<<<1, 64, 0, stream>>>(part, gamma, beta, ss);
    final_kernel<<<256, 256, 0, stream>>>(ft, nn, W, ss, out);
}